// ZAM_Module_17446157156896
// MI455X (gfx1250) — compile-verified
//
#include <hip/hip_runtime.h>
#include <math.h>

// Native 16-byte vector type: legal operand for __builtin_nontemporal_load/store
typedef float v4f __attribute__((ext_vector_type(4)));

// ---------------------------------------------------------------------------
// Kernel 1: compute the Zernike scalar S = sum_{i,j<256} cos(atan2(j,i))
//           * (p+1) / (pi * (N-1)^2), with p=2, q=1, N=256.
// Single block, fixed reduction order -> deterministic. Result -> ws[0].
// ---------------------------------------------------------------------------
__global__ __launch_bounds__(256) void zernike_scale_kernel(float* __restrict__ ws) {
    __shared__ float red[256];
    const int i = threadIdx.x;            // row index i in [0,256)
    float acc = 0.0f;
    #pragma unroll 8
    for (int j = 0; j < 256; ++j) {
        // q = 1: cos(q * theta) = cos(atan2(j, i)); atan2(0,0) == 0 -> cos = 1
        acc += cosf(atan2f((float)j, (float)i));
    }
    red[i] = acc;
    __syncthreads();
    // LDS tree reduction (fixed order, deterministic)
    for (int off = 128; off > 0; off >>= 1) {
        if (i < off) red[i] += red[i + off];
        __syncthreads();
    }
    if (i == 0) {
        const float scale = 3.0f / ((float)M_PI * 255.0f * 255.0f); // (p+1)/(pi*(N-1)^2)
        ws[0] = red[0] * scale;
    }
}

// ---------------------------------------------------------------------------
// Kernel 2: out = gamma * (S * x) + x   (pure HBM-bandwidth-bound stream)
//   - v4f (B128) per access, 4 accesses per thread (MLP)
//   - non-temporal load/store: 256 MiB streamed once, keep L2 clean
//   - one global_prefetch a tile ahead
// Tile layout: block b owns float4 indices [b*1024, (b+1)*1024); thread t
// touches t, t+256, t+512, t+768 -> fully coalesced 128B cachelines.
// ---------------------------------------------------------------------------
#define TPB 256
#define F4_PER_THREAD 4
#define TILE_F4 (TPB * F4_PER_THREAD)   // 1024 float4 = 16 KiB per block

__global__ __launch_bounds__(TPB) void zam_apply_kernel(
    const v4f*   __restrict__ x,
    const float* __restrict__ gamma,
    const float* __restrict__ ws,
    v4f*         __restrict__ out,
    int n4)
{
    const float S = ws[0];        // uniform -> scalar load
    const float g = gamma[0];     // uniform -> scalar load

    const int base = blockIdx.x * TILE_F4 + threadIdx.x;

    // Prefetch the next block-sized tile toward L2 (gfx1250 global_prefetch).
    __builtin_prefetch((const void*)(x + base + TILE_F4), 0, 1);

    v4f v[F4_PER_THREAD];
    int idx[F4_PER_THREAD];

    #pragma unroll
    for (int k = 0; k < F4_PER_THREAD; ++k) {
        idx[k] = base + k * TPB;
        if (idx[k] < n4) {
            v[k] = __builtin_nontemporal_load(&x[idx[k]]);
        }
    }

    #pragma unroll
    for (int k = 0; k < F4_PER_THREAD; ++k) {
        if (idx[k] < n4) {
            // Keep reference op order: gamma * (S * x) + x
            v4f r;
            r.x = g * (S * v[k].x) + v[k].x;
            r.y = g * (S * v[k].y) + v[k].y;
            r.z = g * (S * v[k].z) + v[k].z;
            r.w = g * (S * v[k].w) + v[k].w;
            __builtin_nontemporal_store(r, &out[idx[k]]);
        }
    }
}

// ---------------------------------------------------------------------------
// Launcher
//   d_in[0]: x     (float32, B*C*H*W = 33,554,432 elements, 16B aligned)
//   d_in[1]: gamma (float32, 1 element)
//   d_out  : float32, same shape as x
//   d_ws   : >= 4 bytes scratch for the scalar S
// ---------------------------------------------------------------------------
extern "C" void kernel_launch(void* const* d_in, const int* in_sizes, int n_in,
                              void* d_out, int out_size, void* d_ws, size_t ws_size,
                              hipStream_t stream) {
    const float* x     = (const float*)d_in[0];
    const float* gamma = (const float*)d_in[1];
    float*       ws    = (float*)d_ws;

    const int n  = in_sizes[0];      // total elements (multiple of 4: W=256)
    const int n4 = n / 4;            // float4 count

    // 1) scalar S -> ws[0]  (same stream, in-order: no explicit sync needed)
    zernike_scale_kernel<<<1, 256, 0, stream>>>(ws);

    // 2) streaming elementwise pass
    const int grid = (n4 + TILE_F4 - 1) / TILE_F4;   // 8192 blocks for 33.5M elems
    zam_apply_kernel<<<grid, TPB, 0, stream>>>(
        (const v4f*)x, gamma, ws, (v4f*)d_out, n4);
}